// OmegaRestraint_29231547417077
// MI455X (gfx1250) — compile-verified
//
#include <hip/hip_runtime.h>
#include <cmath>

#define NBINS 24

__device__ __forceinline__ float wave_reduce_add_f32(float v) {
    #pragma unroll
    for (int off = 16; off > 0; off >>= 1)
        v += __shfl_xor(v, off, 32);
    return v;
}

__global__ void __launch_bounds__(256)
omega_restraint_kernel(const float* __restrict__ CA,
                       const float* __restrict__ CB,
                       const float* __restrict__ coeff,
                       const float* __restrict__ cutoffs,
                       const int* __restrict__ x_idx,
                       const int* __restrict__ y_idx,
                       int N, int L, int B,
                       float* __restrict__ partials) {
    __shared__ float s_cut[32];
    __shared__ float s_wsum[8];

    // --- CDNA5 async global->LDS broadcast of the 25-entry cutoffs table ---
    if (threadIdx.x < 25) {
        unsigned lds_addr = (unsigned)(unsigned long long)(&s_cut[threadIdx.x]);
        unsigned goff     = threadIdx.x * 4u;
        asm volatile("global_load_async_to_lds_b32 %0, %1, %2 offset:0"
                     :: "v"(lds_addr), "v"(goff),
                        "s"((unsigned long long)cutoffs)
                     : "memory");
    }
    asm volatile("s_wait_asynccnt 0" ::: "memory");
    __syncthreads();

    const float c0       = s_cut[0];
    const float inv_step = 1.0f / (s_cut[1] - s_cut[0]);
    const float eps2     = 1e-6f * 1e-6f;   // DIH_EPS^2
    const float sc_eps   = 1e-6f;           // DIH_EPS

    float acc = 0.0f;
    const int stride = gridDim.x * blockDim.x;
    int k = blockIdx.x * blockDim.x + threadIdx.x;

    int xk = 0, yk = 0;
    if (k < N) { xk = x_idx[k]; yk = y_idx[k]; }

    while (k < N) {
        // --- software pipeline: load next indices, prefetch next coeff row ---
        const int kn = k + stride;
        int xn = 0, yn = 0;
        if (kn < N) {
            xn = x_idx[kn];
            yn = y_idx[kn];
            const char* pf = (const char*)(coeff +
                ((size_t)xn * (size_t)L + (size_t)yn) * (size_t)(NBINS * 4));
            __builtin_prefetch(pf,       0, 3);   // -> global_prefetch_b8
            __builtin_prefetch(pf + 128, 0, 3);
            __builtin_prefetch(pf + 256, 0, 3);
        }

        const size_t rowBase = ((size_t)xk * (size_t)L + (size_t)yk) * (size_t)NBINS;
        const int xi = xk, yi = yk;

        auto body = [&](int b) {
            const float* ca = CA + (size_t)b * (size_t)L * 3;
            const float* cb = CB + (size_t)b * (size_t)L * 3;

            const float xCA0 = ca[3*xi+0], xCA1 = ca[3*xi+1], xCA2 = ca[3*xi+2];
            const float xCB0 = cb[3*xi+0], xCB1 = cb[3*xi+1], xCB2 = cb[3*xi+2];
            const float yCB0 = cb[3*yi+0], yCB1 = cb[3*yi+1], yCB2 = cb[3*yi+2];
            const float yCA0 = ca[3*yi+0], yCA1 = ca[3*yi+1], yCA2 = ca[3*yi+2];

            const float vx0 = xCB0 - xCA0, vx1 = xCB1 - xCA1, vx2 = xCB2 - xCA2;
            const float vy0 = yCB0 - xCB0, vy1 = yCB1 - xCB1, vy2 = yCB2 - xCB2;
            const float vz0 = yCA0 - yCB0, vz1 = yCA1 - yCB1, vz2 = yCA2 - yCB2;

            const float n2x = vx0*vx0 + vx1*vx1 + vx2*vx2;
            const float n2y = vy0*vy0 + vy1*vy1 + vy2*vy2;
            const float n2z = vz0*vz0 + vz1*vz1 + vz2*vz2;

            // c_xy = cross(x, y)
            const float cx0 = vx1*vy2 - vx2*vy1;
            const float cx1 = vx2*vy0 - vx0*vy2;
            const float cx2 = vx0*vy1 - vx1*vy0;
            // c_yz = cross(y, z)
            const float cz0 = vy1*vz2 - vy2*vz1;
            const float cz1 = vy2*vz0 - vy0*vz2;
            const float cz2 = vy0*vz1 - vy1*vz0;
            // cross(c_xy, c_yz)
            const float cc0 = cx1*cz2 - cx2*cz1;
            const float cc1 = cx2*cz0 - cx0*cz2;
            const float cc2 = cx0*cz1 - cx1*cz0;

            const float sinv = vy0*cc0 + vy1*cc1 + vy2*cc2;
            const float cosv = (cx0*cz0 + cx1*cz1 + cx2*cz2) * sqrtf(n2y);

            const bool valid = (n2x > eps2) && (n2y > eps2) && (n2z > eps2) &&
                               (sinv*sinv + cosv*cosv > sc_eps);
            if (valid) {
                const float omega = atan2f(sinv, cosv);
                int idx = (int)floorf((omega - c0) * inv_step);
                idx = idx < 0 ? 0 : (idx > NBINS - 1 ? NBINS - 1 : idx);
                const float t = omega - s_cut[idx];
                const float4 c = *reinterpret_cast<const float4*>(
                    coeff + (rowBase + (size_t)idx) * 4);
                acc += c.w + t * (c.z + t * (c.y + t * c.x));
            }
        };

        if (B == 2) {
            // Compile-time-unrolled: both batches' coeff gathers go in flight
            // together (2x memory-level parallelism on the random-gather path).
            body(0);
            body(1);
        } else {
            for (int b = 0; b < B; ++b) body(b);
        }

        k = kn; xk = xn; yk = yn;
    }

    // --- deterministic in-block reduction (wave32) ---
    const int lane = threadIdx.x & 31;
    const int wid  = threadIdx.x >> 5;
    const float w = wave_reduce_add_f32(acc);
    if (lane == 0) s_wsum[wid] = w;
    __syncthreads();
    if (wid == 0) {
        float v = (lane < (int)(blockDim.x >> 5)) ? s_wsum[lane] : 0.0f;
        v = wave_reduce_add_f32(v);
        if (lane == 0) partials[blockIdx.x] = v;
    }
}

__global__ void __launch_bounds__(256)
final_reduce_kernel(const float* __restrict__ partials, int n,
                    float* __restrict__ out) {
    __shared__ double s[8];
    double acc = 0.0;
    for (int i = threadIdx.x; i < n; i += blockDim.x)
        acc += (double)partials[i];
    #pragma unroll
    for (int off = 16; off > 0; off >>= 1)
        acc += __shfl_xor(acc, off, 32);
    const int lane = threadIdx.x & 31;
    const int wid  = threadIdx.x >> 5;
    if (lane == 0) s[wid] = acc;
    __syncthreads();
    if (threadIdx.x == 0) {
        double tot = 0.0;
        const int nw = (int)(blockDim.x >> 5);
        for (int i = 0; i < nw; ++i) tot += s[i];
        out[0] = (float)tot;
    }
}

extern "C" void kernel_launch(void* const* d_in, const int* in_sizes, int n_in,
                              void* d_out, int out_size, void* d_ws, size_t ws_size,
                              hipStream_t stream) {
    const float* CA      = (const float*)d_in[0];
    const float* CB      = (const float*)d_in[1];
    const float* coeff   = (const float*)d_in[2];
    const float* cutoffs = (const float*)d_in[3];
    const int*   x_idx   = (const int*)d_in[4];
    const int*   y_idx   = (const int*)d_in[5];

    const int N = in_sizes[4];
    const int coeffN = in_sizes[2];                          // L*L*NBINS*4
    const int L = (int)llround(sqrt((double)coeffN / (double)(NBINS * 4)));
    const int B = (L > 0) ? in_sizes[0] / (L * 3) : 0;

    const int threads = 256;
    int nblocks = (N + threads - 1) / threads;
    if (nblocks < 1) nblocks = 1;
    if (nblocks > 1024) nblocks = 1024;                      // grid-stride: >=2 iters/thread
    const size_t cap = ws_size / sizeof(float);
    if ((size_t)nblocks > cap && cap >= 1) nblocks = (int)cap;

    float* partials = (float*)d_ws;
    omega_restraint_kernel<<<nblocks, threads, 0, stream>>>(
        CA, CB, coeff, cutoffs, x_idx, y_idx, N, L, B, partials);
    final_reduce_kernel<<<1, threads, 0, stream>>>(partials, nblocks, (float*)d_out);
}